// LaConv_12781822673064
// MI455X (gfx1250) — compile-verified
//
#include <hip/hip_runtime.h>
#include <hip/hip_bf16.h>

typedef __attribute__((ext_vector_type(2))) float v2f;
typedef __attribute__((ext_vector_type(4))) float v4f;
typedef __attribute__((ext_vector_type(8))) float v8f;

#define OUTC     72        // K*K*G = 3*3*8
#define MROWS    80        // padded to 5 x 16; rows >= 72 are zeros in LDS
#define MTILES   5
#define CF       512
#define HW       4096      // 64*64
#define KCHUNK   64        // K-slab staged in LDS per iteration
#define LDSTRIDE 66        // 66 floats/row: even (b64-aligned) + bank spread

// ---------------------------------------------------------------------------
// Phase 1: wf[b, o, hw] = sum_c W_lin[o, c] * cond[b, c, hw] + b_lin[o]
// One wave owns one 16-column strip of hw for ALL 80 (padded) output rows:
// the cond (B) fragment is loaded once per K-step and reused by 5 WMMAs.
// W_lin (A) is staged in LDS in 80x64 zero-padded slabs shared by 8 waves,
// so the hot loop has no divergent masked loads and VMEM only streams cond.
// ---------------------------------------------------------------------------
__global__ __launch_bounds__(256) void laconv_wfield_gemm(
    const float* __restrict__ W_lin,   // [72, 512] row-major
    const float* __restrict__ b_lin,   // [72]
    const float* __restrict__ cond,    // [B, 512, 4096]
    float* __restrict__ wf,            // [B, 72, 4096]
    int B)
{
    __shared__ float sA[MROWS * LDSTRIDE];             // 21120 B of 320 KB/WGP

    const int lane = threadIdx.x & 31;
    const int wave = threadIdx.x >> 5;
    const int task = blockIdx.x * 8 + wave;            // one wave = one N-strip
    const int nt_per_batch = HW / 16;                  // 256
    const int b  = task / nt_per_batch;
    const int nt = task % nt_per_batch;
    const int n0   = nt * 16;
    const int half = lane >> 4;                        // 0: lanes 0-15, 1: lanes 16-31
    const int idx  = lane & 15;
    const bool live = (b < B);                         // wave-uniform

    const float* condb = cond + (size_t)b * CF * HW;

    v8f acc[MTILES];
    #pragma unroll
    for (int t = 0; t < MTILES; ++t) { v8f z = {}; acc[t] = z; }

    for (int kc = 0; kc < CF; kc += KCHUNK) {
        __syncthreads();                               // prev slab fully consumed
        // Cooperative stage of A slab: 80 rows x 64 k, rows >= 72 zeroed.
        for (int e = threadIdx.x; e < MROWS * KCHUNK; e += 256) {
            const int row = e >> 6;                    // / KCHUNK
            const int k   = e & (KCHUNK - 1);
            sA[row * LDSTRIDE + k] =
                (row < OUTC) ? W_lin[(size_t)row * CF + kc + k] : 0.f;
        }
        __syncthreads();

        if (live) {
            for (int kk = 0; kk < KCHUNK / 4; ++kk) {
                const int kA = kk * 4 + 2 * half;      // VGPR0->K{0,2}, VGPR1->K{1,3}
                const float* crow = condb + (size_t)(kc + kA) * HW + n0 + idx;
                v2f bfrag;
                bfrag.x = crow[0];
                bfrag.y = crow[HW];
                #pragma unroll
                for (int t = 0; t < MTILES; ++t) {
                    v2f afrag = *(const v2f*)&sA[(t * 16 + idx) * LDSTRIDE + kA];
                    acc[t] = __builtin_amdgcn_wmma_f32_16x16x4_f32(
                        false, afrag, false, bfrag, (short)0, acc[t],
                        /*reuse_a=*/false, /*reuse_b=*/false);
                }
            }
        }
    }

    if (!live) return;
    // D layout: VGPR r, lanes 0-15 -> M=r, lanes 16-31 -> M=r+8; N = lane&15
    #pragma unroll
    for (int t = 0; t < MTILES; ++t) {
        #pragma unroll
        for (int r = 0; r < 8; ++r) {
            const int o = t * 16 + r + 8 * half;
            if (o < OUTC) {
                wf[((size_t)b * OUTC + o) * HW + n0 + idx] = acc[t][r] + b_lin[o];
            }
        }
    }
}

// ---------------------------------------------------------------------------
// Phase 2: out[b, g*64+cg, h, w] = sum_{ki,kj} x[b,c,h+ki-1,w+kj-1] * wf[b,g*9+3ki+kj,h,w]
// 4 outputs per thread along w: b128 loads/stores keep the streaming kernel
// bandwidth-bound, not issue-bound. Out-of-image taps use 0 (== zero padding).
// ---------------------------------------------------------------------------
__global__ __launch_bounds__(256) void laconv_apply(
    const float* __restrict__ x,    // [B, 512, 64, 64]
    const float* __restrict__ wf,   // [B, 72, 4096]
    float* __restrict__ out,
    int total4)
{
    const int i = blockIdx.x * blockDim.x + threadIdx.x;
    if (i >= total4) return;
    const int w0 = (i & 15) << 2;           // 16 float4 groups per row
    const int h  = (i >> 4) & 63;
    const int c  = (i >> 10) & 511;
    const int b  = i >> 19;                 // 512*64*16 = 2^19
    const int g  = c >> 6;                  // Cg = 64

    const float* xc  = x  + ((size_t)(b * 512 + c) << 12);
    const float* wfp = wf + (((size_t)b * OUTC + g * 9) << 12) + (h << 6) + w0;

    v4f acc = {};
    #pragma unroll
    for (int ki = 0; ki < 3; ++ki) {
        const int hh = h + ki - 1;
        if ((unsigned)hh >= 64u) continue;              // zero-padded row
        const float* xrow = xc + (hh << 6);
        const v4f   xv = *(const v4f*)(xrow + w0);
        const float xl = (w0 > 0)  ? xrow[w0 - 1] : 0.f;   // zero padding
        const float xr = (w0 < 60) ? xrow[w0 + 4] : 0.f;   // zero padding
        const v4f s0 = { xl,   xv.x, xv.y, xv.z };      // dw = -1
        const v4f s2 = { xv.y, xv.z, xv.w, xr   };      // dw = +1
        const v4f wv0 = *(const v4f*)(wfp + ((size_t)(ki * 3 + 0) << 12));
        const v4f wv1 = *(const v4f*)(wfp + ((size_t)(ki * 3 + 1) << 12));
        const v4f wv2 = *(const v4f*)(wfp + ((size_t)(ki * 3 + 2) << 12));
        acc += s0 * wv0 + xv * wv1 + s2 * wv2;
    }
    *(v4f*)(out + ((size_t)i << 2)) = acc;
}

extern "C" void kernel_launch(void* const* d_in, const int* in_sizes, int n_in,
                              void* d_out, int out_size, void* d_ws, size_t ws_size,
                              hipStream_t stream) {
    const float* x     = (const float*)d_in[0];   // [B, 512, 64, 64]
    const float* cond  = (const float*)d_in[1];   // [B, 512, 64, 64]
    const float* W_lin = (const float*)d_in[2];   // [72, 512]
    const float* b_lin = (const float*)d_in[3];   // [72]
    float* out = (float*)d_out;
    float* wf  = (float*)d_ws;                    // [B, 72, 4096] = 18.9 MB

    const int B = in_sizes[0] / (512 * 64 * 64);  // 16

    // Phase 1: B*256 N-strips, 8 waves per block
    const int gemm_waves  = B * (HW / 16);
    const int gemm_blocks = (gemm_waves + 7) / 8;
    laconv_wfield_gemm<<<gemm_blocks, 256, 0, stream>>>(W_lin, b_lin, cond, wf, B);

    // Phase 2: one thread per 4 consecutive outputs along w
    const int total4 = B * 512 * 64 * 16;
    laconv_apply<<<(total4 + 255) / 256, 256, 0, stream>>>(x, wf, out, total4);
}